// TransR_62036507623588
// MI455X (gfx1250) — compile-verified
//
#include <hip/hip_runtime.h>
#include <hip/hip_bf16.h>

typedef __attribute__((ext_vector_type(2))) float v2f;
typedef __attribute__((ext_vector_type(8))) float v8f;

#define EDIM 128
#define RDIM 64
#define WAVES 4

// ---------------- helper kernels ----------------

__global__ void zero_kernel(int* p, int n) {
    int i = blockIdx.x * blockDim.x + threadIdx.x;
    if (i < n) p[i] = 0;
}

__global__ void hist_kernel(const int* __restrict__ rel, int* __restrict__ counts, int n) {
    int i = blockIdx.x * blockDim.x + threadIdx.x;
    if (i < n) atomicAdd(&counts[rel[i]], 1);
}

// single-block inclusive->exclusive scan over n (<=1024) counts
__global__ void scan_kernel(const int* __restrict__ counts, int* __restrict__ offsets, int n) {
    __shared__ int tmp[1024];
    int t = threadIdx.x;
    int v = (t < n) ? counts[t] : 0;
    tmp[t] = v;
    __syncthreads();
    for (int off = 1; off < 1024; off <<= 1) {
        int x = (t >= off) ? tmp[t - off] : 0;
        __syncthreads();
        tmp[t] += x;
        __syncthreads();
    }
    if (t < n) offsets[t] = tmp[t] - v;      // exclusive
    if (t == n - 1) offsets[n] = tmp[t];     // total
}

__global__ void scatter_kernel(const int* __restrict__ rel, const int* __restrict__ offsets,
                               int* __restrict__ cursor, int* __restrict__ sortedIdx, int n) {
    int i = blockIdx.x * blockDim.x + threadIdx.x;
    if (i < n) {
        int r = rel[i];
        int p = atomicAdd(&cursor[r], 1);
        sortedIdx[offsets[r] + p] = i;
    }
}

// ---------------- WMMA score kernel ----------------
// One block per relation. 4 waves; each wave processes tiles of 16 triples with
// V_WMMA_F32_16X16X4_F32: D(16x16) += A(16x4) x B(4x16).
// f32 WMMA fragment layouts (ISA 7.12.2): lane = 16*half + col;
//   A: reg0 = A[col][kbase+2*half], reg1 = A[col][kbase+2*half+1]
//   B: reg0 = B[kbase+2*half][nbase+col], reg1 = B[kbase+2*half+1][nbase+col]
//   C/D: reg v = D[v+8*half][col]
// M is staged in LDS k-pair interleaved: float2 at [kpair*64 + n] = {M[2*kpair][n], M[2*kpair+1][n]}
// so every B fragment is one aligned ds_load_b64 straight into a WMMA register pair.
__launch_bounds__(32 * WAVES, 1)
__global__ void transr_score_kernel(const int* __restrict__ head,
                                    const int* __restrict__ tail,
                                    const float* __restrict__ entity,    // [NE][128]
                                    const float* __restrict__ relEmb,    // [NR][64]
                                    const float* __restrict__ relMat,    // [NR][128*64]
                                    const int* __restrict__ offsets,     // [NR+1]
                                    const int* __restrict__ sortedIdx,   // [B]
                                    float* __restrict__ out) {
    __shared__ float sM[EDIM * RDIM];                 // 32 KB, k-pair interleaved (see above)
    __shared__ float sDelta[WAVES][16 * EDIM];        // 32 KB, per-wave [m][k]

    const int r = blockIdx.x;
    const int start = offsets[r];
    const int count = offsets[r + 1] - start;
    if (count == 0) return;

    const int tid  = threadIdx.x;
    const int wave = tid >> 5;
    const int lane = tid & 31;
    const int half = lane >> 4;
    const int col  = lane & 15;

    // cooperative load of M_r into LDS with k-pair interleave.
    // 64 k-pairs x 16 n-quads = 1024 pair-quads; 8 per thread.
    {
        const float4* Mg = reinterpret_cast<const float4*>(relMat + (size_t)r * (EDIM * RDIM));
        float4* sM4 = reinterpret_cast<float4*>(sM);
#pragma unroll
        for (int i = 0; i < 8; ++i) {
            const int p  = tid + 128 * i;
            const int kp = p >> 4;        // k-pair index (k0 = 2*kp)
            const int nq = p & 15;        // n-quad index (n = 4*nq .. 4*nq+3)
            float4 g0 = Mg[(2 * kp) * 16 + nq];       // M[2kp][4nq..4nq+3]
            float4 g1 = Mg[(2 * kp + 1) * 16 + nq];   // M[2kp+1][4nq..4nq+3]
            sM4[kp * 32 + nq * 2]     = make_float4(g0.x, g1.x, g0.y, g1.y);
            sM4[kp * 32 + nq * 2 + 1] = make_float4(g0.z, g1.z, g0.w, g1.w);
        }
    }
    __syncthreads();

    // per-lane relation embedding value for each of the 4 n-tiles (constant per block)
    float relv[4];
#pragma unroll
    for (int nn = 0; nn < 4; ++nn) relv[nn] = relEmb[r * RDIM + nn * 16 + col];

    const int nTiles = (count + 15) >> 4;
    float* sd = &sDelta[wave][0];
    const v2f* sMp = reinterpret_cast<const v2f*>(sM);   // [kpair][n] float2

    for (int tile = wave; tile < nTiles; tile += WAVES) {
        const int base   = start + (tile << 4);
        const int mcount = min(16, count - (tile << 4));

        // stage delta = head_emb - tail_emb for 16 triples into this wave's LDS slab.
        // lane pair (2m, 2m+1) stages row m; each lane does 16 float4.
        {
            const int m   = lane >> 1;
            const int seg = (lane & 1) * 16;
            float4* dst = reinterpret_cast<float4*>(sd + m * EDIM);
            if (m < mcount) {
                const int ti = sortedIdx[base + m];
                const float4* hrow = reinterpret_cast<const float4*>(entity + (size_t)head[ti] * EDIM);
                const float4* trow = reinterpret_cast<const float4*>(entity + (size_t)tail[ti] * EDIM);
#pragma unroll
                for (int i = 0; i < 16; ++i) {
                    float4 h4 = hrow[seg + i];
                    float4 t4 = trow[seg + i];
                    dst[seg + i] = make_float4(h4.x - t4.x, h4.y - t4.y, h4.z - t4.z, h4.w - t4.w);
                }
            } else {
#pragma unroll
                for (int i = 0; i < 16; ++i) dst[seg + i] = make_float4(0.f, 0.f, 0.f, 0.f);
            }
        }
        __builtin_amdgcn_wave_barrier();
        asm volatile("s_wait_dscnt 0" ::: "memory");   // wave-internal LDS store->load fence

        v8f acc[4];
#pragma unroll
        for (int nn = 0; nn < 4; ++nn) acc[nn] = (v8f){0.f, 0.f, 0.f, 0.f, 0.f, 0.f, 0.f, 0.f};

        // A row for this lane: delta[col][kbase + 2*half + {0,1}]  (8B aligned -> ds_load_b64)
        const float* dRow = sd + col * EDIM + 2 * half;
#pragma unroll 8
        for (int kk = 0; kk < 32; ++kk) {
            v2f a = *reinterpret_cast<const v2f*>(dRow + kk * 4);
            const v2f* bRow = sMp + (2 * kk + half) * RDIM + col;   // kpair = 2*kk + half
#pragma unroll
            for (int nn = 0; nn < 4; ++nn) {
                v2f b = bRow[nn * 16];   // single ds_load_b64 -> aligned pair
                acc[nn] = __builtin_amdgcn_wmma_f32_16x16x4_f32(
                    false, a, false, b, (short)0, acc[nn], false, false);
            }
        }

        // diff = delta_proj + rel_emb ; score = sum_n |diff|
        float partial[8];
#pragma unroll
        for (int v = 0; v < 8; ++v) {
            float s = fabsf(acc[0][v] + relv[0]);
            s += fabsf(acc[1][v] + relv[1]);
            s += fabsf(acc[2][v] + relv[2]);
            s += fabsf(acc[3][v] + relv[3]);
            partial[v] = s;
        }
        // reduce across the 16 lanes of each half (offsets <16 stay within half)
#pragma unroll
        for (int v = 0; v < 8; ++v) {
            float s = partial[v];
            s += __shfl_xor(s, 1, 32);
            s += __shfl_xor(s, 2, 32);
            s += __shfl_xor(s, 4, 32);
            s += __shfl_xor(s, 8, 32);
            partial[v] = s;
        }
        if (col == 0) {
#pragma unroll
            for (int v = 0; v < 8; ++v) {
                const int m = v + 8 * half;   // triple row held by this half
                if (m < mcount) out[sortedIdx[base + m]] = partial[v];
            }
        }
        __builtin_amdgcn_wave_barrier();
    }
}

// ---------------- launcher ----------------

extern "C" void kernel_launch(void* const* d_in, const int* in_sizes, int n_in,
                              void* d_out, int out_size, void* d_ws, size_t ws_size,
                              hipStream_t stream) {
    const int*   head = (const int*)d_in[0];
    const int*   rel  = (const int*)d_in[1];
    const int*   tail = (const int*)d_in[2];
    const float* ent  = (const float*)d_in[3];
    const float* re   = (const float*)d_in[4];
    const float* rt   = (const float*)d_in[5];
    float* out = (float*)d_out;

    const int batch = in_sizes[0];
    const int nrel  = in_sizes[4] / RDIM;   // 1000

    int* counts    = (int*)d_ws;            // [1024]
    int* cursor    = counts + 1024;         // [1024]
    int* offsets   = cursor + 1024;         // [1024] (nrel+1 used)
    int* sortedIdx = offsets + 1024;        // [batch]

    const int nblk = (batch + 255) / 256;

    zero_kernel<<<(2048 + 255) / 256, 256, 0, stream>>>(counts, 2048);  // counts + cursor
    hist_kernel<<<nblk, 256, 0, stream>>>(rel, counts, batch);
    scan_kernel<<<1, 1024, 0, stream>>>(counts, offsets, nrel);
    scatter_kernel<<<nblk, 256, 0, stream>>>(rel, offsets, cursor, sortedIdx, batch);
    transr_score_kernel<<<nrel, 32 * WAVES, 0, stream>>>(head, tail, ent, re, rt,
                                                         offsets, sortedIdx, out);
}